// SparseGATLayer_53300544143788
// MI455X (gfx1250) — compile-verified
//
#include <hip/hip_runtime.h>
#include <math.h>

// ---- problem constants (fixed by the reference) ----
#define IN_F   128
#define OUT_F  128
#define ALPHA  0.2f
#define GAT_EPS 1e-16f

typedef __attribute__((ext_vector_type(2))) float v2f;
typedef __attribute__((ext_vector_type(4))) float v4f;
typedef __attribute__((ext_vector_type(8))) float v8f;

// ---------------------------------------------------------------------------
// zero-fill (d_out + segment-sum accumulators must be zeroed every launch)
// ---------------------------------------------------------------------------
__global__ void gat_zero_f32(float* __restrict__ p, int n) {
    int i = blockIdx.x * blockDim.x + threadIdx.x;
    int stride = gridDim.x * blockDim.x;
    for (; i < n; i += stride) p[i] = 0.0f;
}

// ---------------------------------------------------------------------------
// Wh = h @ W  via V_WMMA_F32_16X16X4_F32 (full fp32 matrix path)
// one wave -> one 16-row M tile x full 128-col stripe (8 x v8f accumulators)
//
// W is staged in LDS in a k-pair-interleaved layout so each lane's B fragment
// {W[k][n], W[k+1][n]} is a single aligned ds_load_b64 into an adjacent VGPR
// pair (no repack movs in the WMMA loop):
//   ldsBp[(k/2)*OUT_F + n] = { W[k][n], W[k+1][n] }
// ---------------------------------------------------------------------------
__global__ void gat_gemm_wmma(const float* __restrict__ h,
                              const float* __restrict__ W,
                              float* __restrict__ Wh,
                              int n_nodes) {
    __shared__ v2f ldsBp[(IN_F / 2) * OUT_F];   // 64 KB of the 320 KB WGP LDS
    {
        // coalesced read of W (consecutive lanes -> consecutive n),
        // contiguous 8B writes to LDS
        for (int i = threadIdx.x; i < (IN_F / 2) * OUT_F; i += blockDim.x) {
            const int p = i >> 7;            // k-pair index: k = 2p, 2p+1
            const int n = i & 127;
            v2f b;
            b.x = W[(2 * p) * OUT_F + n];
            b.y = W[(2 * p + 1) * OUT_F + n];
            ldsBp[i] = b;
        }
    }
    __syncthreads();

    const int wave  = threadIdx.x >> 5;          // wave32
    const int lane  = threadIdx.x & 31;
    const int mtile = blockIdx.x * (blockDim.x >> 5) + wave;
    const int mbase = mtile * 16;
    if (mbase >= n_nodes) return;                // wave-uniform exit (EXEC all-1s for WMMA)

    const int half = lane >> 4;                  // 0: lanes 0-15, 1: lanes 16-31
    const int l15  = lane & 15;
    int m = mbase + l15;
    if (m >= n_nodes) m = n_nodes - 1;           // clamp loads; stores guarded below

    v8f acc[8];
#pragma unroll
    for (int nt = 0; nt < 8; ++nt) acc[nt] = (v8f){};

    for (int kc = 0; kc < IN_F / 4; ++kc) {
        const int kA = kc * 4 + half * 2;        // even -> 8B-aligned pair
        const int kp = kc * 2 + half;            // k-pair index (kA/2)
        // A frag: A[m][kA], A[m][kA+1]  (global_load_b64)
        const v2f a = *(const v2f*)(h + (size_t)m * IN_F + kA);
#pragma unroll
        for (int nt = 0; nt < 8; ++nt) {
            const int n = nt * 16 + l15;
            const v2f b = ldsBp[kp * OUT_F + n]; // single ds_load_b64
            acc[nt] = __builtin_amdgcn_wmma_f32_16x16x4_f32(
                /*neg_a=*/false, a, /*neg_b=*/false, b,
                /*c_mod=*/(short)0, acc[nt],
                /*reuse_a=*/false, /*reuse_b=*/false);
        }
    }

    // C/D layout: VGPR v -> M = v + half*8, N = nt*16 + l15
#pragma unroll
    for (int nt = 0; nt < 8; ++nt) {
#pragma unroll
        for (int v = 0; v < 8; ++v) {
            const int row = mbase + v + half * 8;
            if (row < n_nodes)
                Wh[(size_t)row * OUT_F + nt * 16 + l15] = acc[nt][v];
        }
    }
}

// ---------------------------------------------------------------------------
// s1 = Wh @ a1 ; s2 = Wh @ a2   (one wave per node, shuffle reduction)
// ---------------------------------------------------------------------------
__global__ void gat_s1s2(const float* __restrict__ Wh,
                         const float* __restrict__ a,
                         float* __restrict__ s1,
                         float* __restrict__ s2,
                         int n_nodes) {
    const int node = (blockIdx.x * blockDim.x + threadIdx.x) >> 5;
    const int lane = threadIdx.x & 31;
    if (node >= n_nodes) return;
    float p1 = 0.0f, p2 = 0.0f;
#pragma unroll
    for (int f = lane; f < OUT_F; f += 32) {
        const float w = Wh[(size_t)node * OUT_F + f];
        p1 = fmaf(w, a[f], p1);
        p2 = fmaf(w, a[OUT_F + f], p2);
    }
#pragma unroll
    for (int off = 16; off > 0; off >>= 1) {
        p1 += __shfl_down(p1, off, 32);
        p2 += __shfl_down(p2, off, 32);
    }
    if (lane == 0) { s1[node] = p1; s2[node] = p2; }
}

// ---------------------------------------------------------------------------
// edge pass 1: e = leaky_relu(s1[row] + s2[col]); rowsum[row] += e
// ---------------------------------------------------------------------------
__global__ void gat_edge1(const int* __restrict__ row, const int* __restrict__ col,
                          const float* __restrict__ s1, const float* __restrict__ s2,
                          float* __restrict__ e_buf, float* __restrict__ e_rowsum,
                          int n_edges) {
    const int e = blockIdx.x * blockDim.x + threadIdx.x;
    if (e >= n_edges) return;
    const int r = row[e], c = col[e];
    const float v  = s1[r] + s2[c];
    const float ev = v > 0.0f ? v : ALPHA * v;
    e_buf[e] = ev;
    atomicAdd(&e_rowsum[r], ev);
}

// ---------------------------------------------------------------------------
// edge pass 2: ee = exp(e - rowsum[row]); expsum[row] += ee
// ---------------------------------------------------------------------------
__global__ void gat_edge2(const int* __restrict__ row,
                          const float* __restrict__ e_buf,
                          const float* __restrict__ e_rowsum,
                          float* __restrict__ e_exp, float* __restrict__ e_exp_sum,
                          int n_edges) {
    const int e = blockIdx.x * blockDim.x + threadIdx.x;
    if (e >= n_edges) return;
    const int r = row[e];
    const float ee = expf(e_buf[e] - e_rowsum[r]);
    e_exp[e] = ee;
    atomicAdd(&e_exp_sum[r], ee);
}

// ---------------------------------------------------------------------------
// edge pass 3 (heavy, L2-resident gather + atomic scatter):
// out[row][:] += (ee / (expsum[row]+eps)) * Wh[col][:]
// one wave per edge; each lane owns a contiguous float4 of the 128 features.
// ---------------------------------------------------------------------------
__global__ void gat_edge3(const int* __restrict__ row, const int* __restrict__ col,
                          const float* __restrict__ e_exp,
                          const float* __restrict__ e_exp_sum,
                          const float* __restrict__ Wh,
                          float* __restrict__ out,
                          int n_edges) {
    const int e    = (blockIdx.x * blockDim.x + threadIdx.x) >> 5;
    const int lane = threadIdx.x & 31;
    if (e >= n_edges) return;
    const int r = row[e], c = col[e];
    const float att = e_exp[e] / (e_exp_sum[r] + GAT_EPS);
    const v4f w = *(const v4f*)(Wh + (size_t)c * OUT_F + lane * 4);
    float* dst = out + (size_t)r * OUT_F + lane * 4;
    atomicAdd(dst + 0, att * w.x);
    atomicAdd(dst + 1, att * w.y);
    atomicAdd(dst + 2, att * w.z);
    atomicAdd(dst + 3, att * w.w);
}

// ---------------------------------------------------------------------------
extern "C" void kernel_launch(void* const* d_in, const int* in_sizes, int n_in,
                              void* d_out, int out_size, void* d_ws, size_t ws_size,
                              hipStream_t stream) {
    const float* h          = (const float*)d_in[0];
    const int*   edge_index = (const int*)d_in[1];   // (2, E) int32
    const float* W          = (const float*)d_in[2];
    const float* a          = (const float*)d_in[3]; // 256 floats: a1 | a2

    float* out = (float*)d_out;

    const int n_nodes = in_sizes[0] / IN_F;   // 50000
    const int n_edges = in_sizes[1] / 2;      // 800000
    const int* row = edge_index;
    const int* col = edge_index + n_edges;

    // workspace carve-up (~33 MB)
    float* ws        = (float*)d_ws;
    float* Wh        = ws;  ws += (size_t)n_nodes * OUT_F;
    float* s1        = ws;  ws += n_nodes;
    float* s2        = ws;  ws += n_nodes;
    float* e_rowsum  = ws;  ws += n_nodes;
    float* e_exp_sum = ws;  ws += n_nodes;
    float* e_buf     = ws;  ws += n_edges;
    float* e_exp     = ws;  ws += n_edges;

    // 0) zero accumulators + output (harness poisons, never re-zeros)
    gat_zero_f32<<<2048, 256, 0, stream>>>(out, n_nodes * OUT_F);
    gat_zero_f32<<<256,  256, 0, stream>>>(e_rowsum, n_nodes);
    gat_zero_f32<<<256,  256, 0, stream>>>(e_exp_sum, n_nodes);

    // 1) Wh = h @ W  (fp32 WMMA)
    const int mtiles = (n_nodes + 15) / 16;            // 3125
    const int waves_per_block = 8;                     // 256 threads = 8 wave32
    gat_gemm_wmma<<<(mtiles + waves_per_block - 1) / waves_per_block,
                    waves_per_block * 32, 0, stream>>>(h, W, Wh, n_nodes);

    // 2) s1, s2
    gat_s1s2<<<((size_t)n_nodes * 32 + 255) / 256, 256, 0, stream>>>(Wh, a, s1, s2, n_nodes);

    // 3-5) edge passes
    gat_edge1<<<(n_edges + 255) / 256, 256, 0, stream>>>(row, col, s1, s2, e_buf, e_rowsum, n_edges);
    gat_edge2<<<(n_edges + 255) / 256, 256, 0, stream>>>(row, e_buf, e_rowsum, e_exp, e_exp_sum, n_edges);
    gat_edge3<<<((size_t)n_edges * 32 + 255) / 256, 256, 0, stream>>>(row, col, e_exp, e_exp_sum, Wh, out, n_edges);
}